// SelfMixing_14516989461263
// MI455X (gfx1250) — compile-verified
//
#include <hip/hip_runtime.h>

// ============================================================================
// Compile-time port of the reference's _su2_cg / _q / _real_cg (double prec).
// ============================================================================
namespace cgc {

constexpr double cfact(int n) { double r = 1.0; for (int i = 2; i <= n; ++i) r *= (double)i; return r; }
constexpr double cabs(double x) { return x < 0.0 ? -x : x; }
constexpr double csqrt(double x) {
  if (x <= 0.0) return 0.0;
  double g = x > 1.0 ? x : 1.0;
  for (int i = 0; i < 64; ++i) g = 0.5 * (g + x / g);
  return g;
}

struct SU2 { double v[5][5][7]; };   // [m1+j1][m2+j2][m3+j3], j1,j2<=2, j3<=3
constexpr SU2 su2_cg(int j1, int j2, int j3) {
  SU2 C{};
  const double pref_j = csqrt((2.0 * j3 + 1.0) * cfact(j3 + j1 - j2) * cfact(j3 - j1 + j2) *
                              cfact(j1 + j2 - j3) / cfact(j1 + j2 + j3 + 1));
  for (int m1 = -j1; m1 <= j1; ++m1)
    for (int m2 = -j2; m2 <= j2; ++m2) {
      const int m3 = m1 + m2;
      if (m3 < -j3 || m3 > j3) continue;
      const double pref_m = csqrt(cfact(j3 + m3) * cfact(j3 - m3) * cfact(j1 - m1) *
                                  cfact(j1 + m1) * cfact(j2 - m2) * cfact(j2 + m2));
      double s = 0.0;
      for (int v = 0; v <= j1 + j2 - j3; ++v) {
        const int a = j1 + j2 - j3 - v, b = j1 - m1 - v, c = j2 + m2 - v;
        const int d = j3 - j2 + m1 + v, e = j3 - j1 - m2 + v;
        if (a < 0 || b < 0 || c < 0 || d < 0 || e < 0) continue;
        const double t = 1.0 / (cfact(v) * cfact(a) * cfact(b) * cfact(c) * cfact(d) * cfact(e));
        s += (v % 2 == 0) ? t : -t;
      }
      C.v[m1 + j1][m2 + j2][m3 + j3] = pref_j * pref_m * s;
    }
  return C;
}

struct QM { double re[7][7]; double im[7][7]; };  // complex->real basis change, l<=3
constexpr QM qmat(int l) {
  QM q{};
  const double s2 = 1.0 / csqrt(2.0);
  for (int m = -l; m < 0; ++m) {
    q.re[l + m][l - m] = s2;    // q[l+m, l+|m|] =  s2
    q.im[l + m][l + m] = -s2;   // q[l+m, l-|m|] = -i*s2
  }
  q.re[l][l] = 1.0;
  for (int m = 1; m <= l; ++m) {
    const double sg = (m % 2 == 0) ? 1.0 : -1.0;
    q.re[l + m][l + m] = sg * s2;   // (-1)^m * s2
    q.im[l + m][l - m] = sg * s2;   // i*(-1)^m * s2
  }
  // multiply by (-i)^l
  double pr = 1.0, pi = 0.0;
  const int ph = l % 4;
  if (ph == 1) { pr = 0.0; pi = -1.0; }
  else if (ph == 2) { pr = -1.0; pi = 0.0; }
  else if (ph == 3) { pr = 0.0; pi = 1.0; }
  for (int a = 0; a < 2 * l + 1; ++a)
    for (int b = 0; b < 2 * l + 1; ++b) {
      const double re = q.re[a][b], im = q.im[a][b];
      q.re[a][b] = re * pr - im * pi;
      q.im[a][b] = re * pi + im * pr;
    }
  return q;
}

struct RCG { double v[5][5][7]; };  // real CG: [i over 2l1+1][j over 2l2+1][k over 2l3+1]
constexpr RCG real_cg(int l1, int l2, int l3) {
  const SU2 C = su2_cg(l1, l2, l3);
  const QM q1 = qmat(l1), q2 = qmat(l2), q3 = qmat(l3);
  RCG out{};
  double Rre[5][5][7] = {}, Rim[5][5][7] = {};
  double sre = 0.0, sim = 0.0;
  const int n1 = 2 * l1 + 1, n2 = 2 * l2 + 1, n3 = 2 * l3 + 1;
  for (int i = 0; i < n1; ++i)
    for (int j = 0; j < n2; ++j)
      for (int k = 0; k < n3; ++k) {
        double ar = 0.0, ai = 0.0;
        for (int a = 0; a < n1; ++a) {
          const double q1r = q1.re[a][i], q1i = q1.im[a][i];
          if (q1r == 0.0 && q1i == 0.0) continue;
          for (int b = 0; b < n2; ++b) {
            const double q2r = q2.re[b][j], q2i = q2.im[b][j];
            if (q2r == 0.0 && q2i == 0.0) continue;
            const double pr = q1r * q2r - q1i * q2i;
            const double pi = q1r * q2i + q1i * q2r;
            for (int c = 0; c < n3; ++c) {
              const double Cv = C.v[a][b][c];
              if (Cv == 0.0) continue;
              const double q3r = q3.re[c][k], q3i = -q3.im[c][k];  // conj
              ar += Cv * (pr * q3r - pi * q3i);
              ai += Cv * (pr * q3i + pi * q3r);
            }
          }
        }
        Rre[i][j][k] = ar; Rim[i][j][k] = ai;
        sre += cabs(ar); sim += cabs(ai);
      }
  const bool useRe = (sre >= sim);
  for (int i = 0; i < n1; ++i)
    for (int j = 0; j < n2; ++j)
      for (int k = 0; k < n3; ++k)
        out.v[i][j][k] = useRe ? Rre[i][j][k] : Rim[i][j][k];
  return out;
}

// One top-level constexpr per table => separate constexpr evaluations (step limit).
constexpr RCG R000 = real_cg(0,0,0);
constexpr RCG R110 = real_cg(1,1,0);
constexpr RCG R220 = real_cg(2,2,0);
constexpr RCG R011 = real_cg(0,1,1);
constexpr RCG R101 = real_cg(1,0,1);
constexpr RCG R111 = real_cg(1,1,1);
constexpr RCG R121 = real_cg(1,2,1);
constexpr RCG R211 = real_cg(2,1,1);
constexpr RCG R221 = real_cg(2,2,1);
constexpr RCG R022 = real_cg(0,2,2);
constexpr RCG R112 = real_cg(1,1,2);
constexpr RCG R122 = real_cg(1,2,2);
constexpr RCG R202 = real_cg(2,0,2);
constexpr RCG R212 = real_cg(2,1,2);
constexpr RCG R222 = real_cg(2,2,2);
constexpr RCG R123 = real_cg(1,2,3);
constexpr RCG R213 = real_cg(2,1,3);
constexpr RCG R223 = real_cg(2,2,3);

// Couplings in the reference's enumeration order: for lo, for l1, for l2.
struct CInfo { int lo, l1, l2; };
constexpr CInfo CPL[18] = {
  {0,0,0},{0,1,1},{0,2,2},
  {1,0,1},{1,1,0},{1,1,1},{1,1,2},{1,2,1},{1,2,2},
  {2,0,2},{2,1,1},{2,1,2},{2,2,0},{2,2,1},{2,2,2},
  {3,1,2},{3,2,1},{3,2,2}};
// CG table for coupling c is real_cg(l1, l2, lo):
constexpr RCG RT[18] = { R000, R110, R220,
                         R011, R101, R111, R121, R211, R221,
                         R022, R112, R122, R202, R212, R222,
                         R123, R213, R223 };

constexpr int MOFF[4] = {0, 1, 4, 9};  // prefix sums of (2l+1): component offsets

struct Ent { signed char m1, m2, kl; float v; };  // global m-indices, local k, 0.5*cg
struct Tab { int n; int coff[19]; int nk[18]; int ko[18]; Ent e[320]; };

constexpr Tab build() {
  Tab t{};
  int n = 0;
  for (int c = 0; c < 18; ++c) {
    t.coff[c] = n;
    const int lo = CPL[c].lo, l1 = CPL[c].l1, l2 = CPL[c].l2;
    t.nk[c] = 2 * lo + 1;
    t.ko[c] = MOFF[lo];
    for (int i = 0; i < 2 * l1 + 1; ++i)
      for (int j = 0; j < 2 * l2 + 1; ++j)
        for (int k = 0; k < 2 * lo + 1; ++k) {
          const double v = RT[c].v[i][j][k];
          if (cabs(v) > 1e-9) {
            t.e[n].m1 = (signed char)(MOFF[l1] + i);
            t.e[n].m2 = (signed char)(MOFF[l2] + j);
            t.e[n].kl = (signed char)k;
            t.e[n].v  = (float)(0.5 * v);
            ++n;
          }
        }
  }
  t.coff[18] = n;
  t.n = n;
  return t;
}
constexpr Tab TB = build();
static_assert(TB.n >= 100 && TB.n <= 320, "unexpected CG sparsity");

} // namespace cgc

// ============================================================================
// Kernel
// ============================================================================
constexpr int NCH  = 128;
constexpr int DIN  = 1152;   // (1+3+5)*128
constexpr int DOUT = 2048;   // (1+3+5+7)*128
constexpr int G    = 8;      // batch rows per thread

__device__ __forceinline__ void async_g2l_b128(unsigned lds_byte, const void* gbase, int goff) {
  // CDNA5 async copy global -> LDS (GVS mode: saddr base + vgpr byte offset).
  asm volatile("global_load_async_to_lds_b128 %0, %1, %2"
               :: "v"(lds_byte), "v"(goff), "s"(gbase) : "memory");
}

__global__ __launch_bounds__(256) void selfmix_kernel(
    const float* __restrict__ x, const float* __restrict__ keep,
    const float* __restrict__ mix, float* __restrict__ out, int nrows)
{
  using namespace cgc;
  __shared__ float scoef[(3 + 18) * NCH];  // [0,384): keep, [384,2688): mix

  const int tid = threadIdx.x;
  // ---- Stage coefficient tables into LDS via the CDNA5 async path ----
  const unsigned lds0 = (unsigned)(size_t)(&scoef[0]);
  if (tid < 96)                                  // 384 floats  = 96 x b128
    async_g2l_b128(lds0 + (unsigned)tid * 16u, keep, tid * 16);
  #pragma unroll
  for (int u = 0; u < 3; ++u) {                  // 2304 floats = 576 x b128
    const int idx = tid + u * 256;
    if (idx < 576)
      async_g2l_b128(lds0 + 1536u + (unsigned)idx * 16u, mix, idx * 16);
  }
  asm volatile("s_wait_asynccnt 0" ::: "memory");
  __syncthreads();

  const int n  = tid & (NCH - 1);
  const int rg = blockIdx.x * 2 + (tid >> 7);    // row-group index

  float keepr[3], mixr[18];
  #pragma unroll
  for (int l = 0; l < 3; ++l)  keepr[l] = scoef[l * NCH + n];
  #pragma unroll
  for (int c = 0; c < 18; ++c) mixr[c] = scoef[384 + c * NCH + n];

  const size_t base = (size_t)rg * G;
  #pragma unroll 1
  for (int it = 0; it < G; ++it) {
    const size_t r = base + (size_t)it;
    if (r >= (size_t)nrows) break;
    const float* xp = x + r * DIN + n;

    if (it + 1 < G && r + 1 < (size_t)nrows) {   // gfx1250 global_prefetch next row
      #pragma unroll
      for (int m = 0; m < 9; ++m) __builtin_prefetch(xp + DIN + m * NCH, 0, 1);
    }

    float xv[9];
    #pragma unroll
    for (int m = 0; m < 9; ++m) xv[m] = __builtin_nontemporal_load(xp + m * NCH);

    float acc[16];
    acc[0] = xv[0] * keepr[0];
    #pragma unroll
    for (int m = 0; m < 3; ++m) acc[1 + m] = xv[1 + m] * keepr[1];
    #pragma unroll
    for (int m = 0; m < 5; ++m) acc[4 + m] = xv[4 + m] * keepr[2];
    #pragma unroll
    for (int k = 9; k < 16; ++k) acc[k] = 0.0f;

    // Fully unrolled sparse CG contraction: all indices/values fold to constants.
    #pragma unroll
    for (int c = 0; c < 18; ++c) {
      float tmp[7];
      #pragma unroll
      for (int k = 0; k < 7; ++k) tmp[k] = 0.0f;
      #pragma unroll
      for (int e = TB.coff[c]; e < TB.coff[c + 1]; ++e)
        tmp[TB.e[e].kl] = fmaf(TB.e[e].v, xv[TB.e[e].m1] * xv[TB.e[e].m2], tmp[TB.e[e].kl]);
      #pragma unroll
      for (int k = 0; k < 7; ++k)
        if (k < TB.nk[c])
          acc[TB.ko[c] + k] = fmaf(mixr[c], tmp[k], acc[TB.ko[c] + k]);
    }

    float* op = out + r * DOUT + n;
    #pragma unroll
    for (int k = 0; k < 16; ++k) __builtin_nontemporal_store(acc[k], op + k * NCH);
  }
}

// ============================================================================
// Launch
// ============================================================================
extern "C" void kernel_launch(void* const* d_in, const int* in_sizes, int n_in,
                              void* d_out, int out_size, void* d_ws, size_t ws_size,
                              hipStream_t stream) {
  (void)n_in; (void)out_size; (void)d_ws; (void)ws_size;
  const float* x    = (const float*)d_in[0];
  const float* keep = (const float*)d_in[1];
  const float* mix  = (const float*)d_in[2];
  float* out = (float*)d_out;

  const int nrows = in_sizes[0] / DIN;                 // 16384
  const int rgs = (nrows + G - 1) / G;                 // row-groups of G rows
  const int blocks = (rgs + 1) / 2;                    // 2 row-groups per 256-thr block
  selfmix_kernel<<<blocks, 256, 0, stream>>>(x, keep, mix, out, nrows);
}